// TPOLoss_47794396070464
// MI455X (gfx1250) — compile-verified
//
#include <hip/hip_runtime.h>
#include <hip/hip_bf16.h>

// MI455X / gfx1250 TPO loss.
// Memory-bound: hidden_state is 1 GiB f32 -> ~46us floor at 23.3 TB/s.
// Segment-sum (onehot^T @ hidden) done with V_WMMA_F32_16X16X4_F32 (exact f32).

typedef float v2f __attribute__((ext_vector_type(2)));
typedef float v8f __attribute__((ext_vector_type(8)));

#define Bdim   8
#define Ndim   8
#define Tdim   2048
#define Hdim   2048
#define NSTEP  16
#define HB     128    // h-columns per block
#define WAVET  256    // t-range per wave (8 waves x 256 = 2048)
#define EPSV   1e-8f
#define BETAV  0.1f

__device__ __forceinline__ float log_sigmoidf(float x) {
    // stable: log sigma(x) = min(x,0) - log1p(exp(-|x|))
    return fminf(x, 0.0f) - log1pf(expf(-fabsf(x)));
}

// ---------------------------------------------------------------------------
// Kernel 1: S[bn][s][h] = sum_t (step_index[bn][t]==s) * hidden[bn][t][h]
// grid.x = 64 (bn) * 16 (h-chunks) = 1024, block = 256 (8 waves)
// ---------------------------------------------------------------------------
__global__ __launch_bounds__(256) void k_segsum(const float* __restrict__ hidden,
                                                const int* __restrict__ stepidx,
                                                float* __restrict__ S) {
    __shared__ float acc[NSTEP][HB];                     // 8 KB
    const int bn = blockIdx.x >> 4;
    const int hb = (blockIdx.x & 15) * HB;
    const int tid = threadIdx.x;

    for (int i = tid; i < NSTEP * HB; i += 256) ((float*)acc)[i] = 0.0f;
    __syncthreads();

    const int wave = tid >> 5;
    const int lane = tid & 31;
    const int m    = lane & 15;
    const bool lo  = lane < 16;
    const int klo  = lo ? 0 : 2;                         // K of B-vgpr0 / A-vgpr0

    const float* hbase = hidden + (size_t)bn * Tdim * Hdim + hb + m;
    const int*   sbase = stepidx + bn * Tdim + wave * WAVET;

    v8f c[8];
    const v8f vzero = {0.f,0.f,0.f,0.f,0.f,0.f,0.f,0.f};
#pragma unroll
    for (int j = 0; j < 8; ++j) c[j] = vzero;

    for (int tc = 0; tc < WAVET; tc += 4) {
        const int4 sv = *(const int4*)(sbase + tc);      // 4 wave-uniform step ids
        v2f a;
        a.x = (m == (lo ? sv.x : sv.z)) ? 1.0f : 0.0f;   // A: K = lo?0:2
        a.y = (m == (lo ? sv.y : sv.w)) ? 1.0f : 0.0f;   // A: K = lo?1:3
        const float* hrow = hbase + (size_t)(wave * WAVET + tc + klo) * Hdim;
#pragma unroll
        for (int j = 0; j < 8; ++j) {                    // 8 h-tiles of 16
            v2f b;
            b.x = hrow[j * 16];                          // B: K = lo?0:2, N = m
            b.y = hrow[Hdim + j * 16];                   // B: K = lo?1:3, N = m
            c[j] = __builtin_amdgcn_wmma_f32_16x16x4_f32(
                false, a, false, b, (short)0, c[j], false, false);
        }
    }

    // C layout: vgpr r of lane L -> row s = r + (L<16?0:8), col h = j*16 + (L&15)
    const int srow = lo ? 0 : 8;
#pragma unroll
    for (int j = 0; j < 8; ++j)
#pragma unroll
        for (int r = 0; r < 8; ++r)
            atomicAdd(&acc[srow + r][j * 16 + m], c[j][r]);
    __syncthreads();

    float* out = S + (size_t)bn * NSTEP * Hdim + hb;
    for (int i = tid; i < NSTEP * HB; i += 256) {
        const int s = i / HB, h = i % HB;
        out[(size_t)s * Hdim + h] = acc[s][h];
    }
}

// ---------------------------------------------------------------------------
// Kernel 2: per-(b,n) step counts + logit segment sums. grid = 64 blocks.
// ---------------------------------------------------------------------------
__global__ __launch_bounds__(256) void k_counts(const int* __restrict__ stepidx,
                                                const float* __restrict__ pol,
                                                const float* __restrict__ ref,
                                                float* __restrict__ cnt,
                                                float* __restrict__ lsum) {
    __shared__ float c[NSTEP], l[NSTEP];
    const int bn = blockIdx.x;
    const int tid = threadIdx.x;
    if (tid < NSTEP) { c[tid] = 0.0f; l[tid] = 0.0f; }
    __syncthreads();
    for (int t = tid; t < Tdim; t += 256) {
        const int idx = bn * Tdim + t;
        const int s = stepidx[idx];
        atomicAdd(&c[s], 1.0f);
        atomicAdd(&l[s], pol[idx] - ref[idx]);
    }
    __syncthreads();
    if (tid < NSTEP) {
        cnt[bn * NSTEP + tid]  = c[tid];
        lsum[bn * NSTEP + tid] = l[tid];
    }
}

// ---------------------------------------------------------------------------
// Kernel 3: dot & norm^2 of step-means vs the n==0 reference means.
// grid.x = 64*16 = 1024 (bn, s), block = 256.
// ---------------------------------------------------------------------------
__global__ __launch_bounds__(256) void k_dots(const float* __restrict__ S,
                                              const float* __restrict__ cnt,
                                              float* __restrict__ dotv,
                                              float* __restrict__ nrm2) {
    const int bn = blockIdx.x >> 4;
    const int s  = blockIdx.x & 15;
    const int b  = bn >> 3;
    const int tid = threadIdx.x;
    const float sc  = 1.0f / fmaxf(cnt[bn * NSTEP + s], 1.0f);
    const float scr = 1.0f / fmaxf(cnt[(b * Ndim) * NSTEP + s], 1.0f);
    const float* x = S + ((size_t)bn * NSTEP + s) * Hdim;
    const float* r = S + ((size_t)(b * Ndim) * NSTEP + s) * Hdim;
    float d = 0.0f, n = 0.0f;
    for (int h = tid; h < Hdim; h += 256) {
        const float xv = x[h] * sc;
        const float rv = r[h] * scr;
        d += xv * rv;
        n += xv * xv;
    }
    __shared__ float sd[256], sn[256];
    sd[tid] = d; sn[tid] = n; __syncthreads();
    for (int off = 128; off > 0; off >>= 1) {
        if (tid < off) { sd[tid] += sd[tid + off]; sn[tid] += sn[tid + off]; }
        __syncthreads();
    }
    if (tid == 0) {
        dotv[bn * NSTEP + s] = sd[0];
        nrm2[bn * NSTEP + s] = sn[0];
    }
}

// ---------------------------------------------------------------------------
// Kernel 4: cos weights -> text logits -> lambda-rank -> loss; reward means.
// 1 block, 256 threads.
// ---------------------------------------------------------------------------
__global__ __launch_bounds__(256) void k_final(const float* __restrict__ cnt,
                                               const float* __restrict__ lsum,
                                               const float* __restrict__ dotv,
                                               const float* __restrict__ nrm2,
                                               const float* __restrict__ labels,
                                               const float* __restrict__ pol,
                                               const float* __restrict__ ref,
                                               float* __restrict__ out) {
    __shared__ float text[Bdim * Ndim];
    __shared__ float red[256];
    const int tid = threadIdx.x;

    if (tid < Bdim * Ndim) {
        const int bn = tid, b = bn >> 3;
        float wsum = 0.0f, tw = 0.0f;
        for (int s = 1; s < NSTEP; ++s) {
            const float cn = cnt[bn * NSTEP + s];
            const float cr = cnt[(b * Ndim) * NSTEP + s];
            if (cn > 0.0f && cr > 0.0f) {
                const float nr = fmaxf(sqrtf(nrm2[bn * NSTEP + s]), EPSV);
                const float rr = fmaxf(sqrtf(nrm2[(b * Ndim) * NSTEP + s]), EPSV);
                const float cosv = dotv[bn * NSTEP + s] / (nr * rr);
                const float w = cosv + 1.0f;
                wsum += w * (lsum[bn * NSTEP + s] / fmaxf(cn, 1.0f));
                tw   += w;
            }
        }
        text[bn] = (tw > 0.0f) ? wsum / fmaxf(tw, EPSV) : 0.0f;
    }
    __syncthreads();

    // lambda-rank: thread (b,i) sums over j (includes i==j -> log 2, as in ref)
    float p = 0.0f;
    if (tid < Bdim * Ndim) {
        const int b = tid >> 3, i = tid & 7;
        const float ti = text[b * Ndim + i];
        const float li = labels[b * Ndim + i];
        for (int j = 0; j < Ndim; ++j) {
            const float diff = ti - text[b * Ndim + j];
            const float ld = li - labels[b * Ndim + j];
            const float sg = (ld > 0.0f) ? 1.0f : (ld < 0.0f ? -1.0f : 0.0f);
            p += -log_sigmoidf(diff * sg);
        }
    }
    red[tid] = p; __syncthreads();
    for (int off = 128; off > 0; off >>= 1) {
        if (tid < off) red[tid] += red[tid + off];
        __syncthreads();
    }
    if (tid == 0) {
        const float lr = red[0] / (8.0f * (float)(Ndim * (Ndim - 1)));
        out[0] = -log_sigmoidf(BETAV * lr);
    }
    __syncthreads();

    // chosen (n=0) / rejected (n=N-1) reward means over (B,T)
    float s0 = 0.0f, s1 = 0.0f;
    for (int e = tid; e < Bdim * Tdim; e += 256) {
        const int b = e / Tdim, t = e % Tdim;
        const int i0 = (b * Ndim + 0) * Tdim + t;
        const int i1 = (b * Ndim + (Ndim - 1)) * Tdim + t;
        s0 += pol[i0] - ref[i0];
        s1 += pol[i1] - ref[i1];
    }
    red[tid] = s0; __syncthreads();
    for (int off = 128; off > 0; off >>= 1) {
        if (tid < off) red[tid] += red[tid + off];
        __syncthreads();
    }
    if (tid == 0) out[1] = red[0] / (float)(Bdim * Tdim);
    __syncthreads();
    red[tid] = s1; __syncthreads();
    for (int off = 128; off > 0; off >>= 1) {
        if (tid < off) red[tid] += red[tid + off];
        __syncthreads();
    }
    if (tid == 0) out[2] = red[0] / (float)(Bdim * Tdim);
}

// ---------------------------------------------------------------------------
extern "C" void kernel_launch(void* const* d_in, const int* in_sizes, int n_in,
                              void* d_out, int out_size, void* d_ws, size_t ws_size,
                              hipStream_t stream) {
    const float* pol    = (const float*)d_in[0];  // (8,8,2048,1)
    const float* refp   = (const float*)d_in[1];  // (8,8,2048,1)
    const float* hid    = (const float*)d_in[2];  // (8,8,2048,2048)
    const int*   sidx   = (const int*)d_in[3];    // (8,8,2048)
    const float* labels = (const float*)d_in[4];  // (8,8)
    float* out = (float*)d_out;                   // 3 f32: loss, chosen, rejected

    float* ws   = (float*)d_ws;
    float* S    = ws;                                   // 64*16*2048 = 2,097,152 f32
    float* cnt  = ws + (size_t)64 * NSTEP * Hdim;       // 1024
    float* ls   = cnt + 64 * NSTEP;                     // 1024
    float* dv   = ls  + 64 * NSTEP;                     // 1024
    float* n2   = dv  + 64 * NSTEP;                     // 1024

    k_segsum<<<dim3(1024), dim3(256), 0, stream>>>(hid, sidx, S);
    k_counts<<<dim3(64),   dim3(256), 0, stream>>>(sidx, pol, refp, cnt, ls);
    k_dots  <<<dim3(1024), dim3(256), 0, stream>>>(S, cnt, dv, n2);
    k_final <<<dim3(1),    dim3(256), 0, stream>>>(cnt, ls, dv, n2, labels, pol, refp, out);
}